// MPL_14568529068457
// MI455X (gfx1250) — compile-verified
//
#include <hip/hip_runtime.h>
#include <math.h>

typedef __attribute__((ext_vector_type(2))) float v2f;
typedef __attribute__((ext_vector_type(8))) float v8f;

#define CLS      10
#define PPC      5
#define NPX      50          // CLS*PPC real proxy rows
#define NPAD     64          // padded proxy rows
#define DSTR     64          // dis row stride (padded, branch-free stores)
#define DIM      256
#define INV_T    20.0f       // 1/TEMP == 1/EPS
#define NB_RED   256         // reduction blocks for row passes

// ---------------------------------------------------------------------------
// K0: build zero-padded proxy copy (64 x 256)
// ---------------------------------------------------------------------------
__global__ void setup_kernel(const float* __restrict__ proxy,
                             float* __restrict__ pp) {
    int r = blockIdx.x;                              // 0..63
    for (int k = threadIdx.x; k < DIM; k += blockDim.x)
        pp[r * DIM + k] = (r < NPX) ? proxy[r * DIM + k] : 0.0f;
}

// ---------------------------------------------------------------------------
// K1: dis = clip(normalize(f) @ proxy.T, -10, 10)  via V_WMMA_F32_16X16X4_F32
// block = 128 threads (4 waves); each block: 16 feature rows; wave w covers
// proxy columns [16w, 16w+16).  dis stored with stride 64 (pad cols are 0).
// ---------------------------------------------------------------------------
#define SA_STRIDE 258   // stride%64 = 2 -> conflict-free row access

__global__ __launch_bounds__(128) void dis_wmma_kernel(
        const float* __restrict__ features,
        const float* __restrict__ pp,          // padded proxy 64x256
        float* __restrict__ dis,               // N x 64
        int N) {
    __shared__ float sA[16 * SA_STRIDE];
    __shared__ float nred[128];
    __shared__ float invn[16];

    const int  tid = threadIdx.x;
    const long r0  = (long)blockIdx.x * 16;

    // ---- stage A tile (16 x 256) into LDS, coalesced float4 ----
    for (int slot = tid; slot < 1024; slot += 128) {
        int row = slot >> 6;               // 64 float4 per row
        int c4  = slot & 63;
        float4 v = make_float4(0.f, 0.f, 0.f, 0.f);
        long gr = r0 + row;
        if (gr < (long)N)
            v = *(const float4*)(features + gr * DIM + c4 * 4);
        float* d = &sA[row * SA_STRIDE + c4 * 4];
        d[0] = v.x; d[1] = v.y; d[2] = v.z; d[3] = v.w;
    }
    __syncthreads();

    // ---- row norms (raw features): 8 threads per row, 32 elems each ----
    {
        int row = tid >> 3, seg = tid & 7;
        const float* p = &sA[row * SA_STRIDE + seg * 32];
        float s = 0.f;
#pragma unroll
        for (int i = 0; i < 32; ++i) { float x = p[i]; s += x * x; }
        nred[tid] = s;
    }
    __syncthreads();
    if (tid < 16) {
        float s = 0.f;
#pragma unroll
        for (int k = 0; k < 8; ++k) s += nred[tid * 8 + k];
        invn[tid] = 1.0f / fmaxf(sqrtf(s), 1e-12f);
    }
    __syncthreads();

    // ---- WMMA: 16x16 tile per wave, K = 256 in 64 steps of 4 ----
    const int wave = tid >> 5;
    const int lane = tid & 31;
    const int mrow = lane & 15;
    const int koff = (lane >> 4) << 1;     // lanes 0-15: K {0,1}; 16-31: K {2,3}
    const int j0   = wave * 16;
    const int jcol = j0 + mrow;

    const float* aPtr = &sA[mrow * SA_STRIDE + koff];
    const float* bPtr = pp + (long)jcol * DIM + koff;

    v8f acc = {};
#pragma unroll 8
    for (int s = 0; s < 64; ++s) {
        v2f a = *(const v2f*)(aPtr + 4 * s);   // ds_load_b64
        v2f b = *(const v2f*)(bPtr + 4 * s);   // global_load_b64 (cache-hot)
        acc = __builtin_amdgcn_wmma_f32_16x16x4_f32(
                false, a, false, b, (short)0, acc, false, false);
    }

    // ---- scale by 1/||f||, clip, store (C layout: VGPR i -> M=i+8*(lane>=16)) ----
    const int rshift = (lane >> 4) << 3;   // 0 or 8
    const int col    = j0 + (lane & 15);
    const float4* ivp = (const float4*)&invn[rshift];
    float4 iv0 = ivp[0], iv1 = ivp[1];
    float ivr[8] = { iv0.x, iv0.y, iv0.z, iv0.w, iv1.x, iv1.y, iv1.z, iv1.w };

    if (r0 + 16 <= (long)N) {
        float* outp = dis + (r0 + rshift) * DSTR + col;
#pragma unroll
        for (int i = 0; i < 8; ++i) {
            float v = acc[i] * ivr[i];
            outp[i * DSTR] = fminf(fmaxf(v, -10.0f), 10.0f);
        }
    } else {
#pragma unroll
        for (int i = 0; i < 8; ++i) {
            long gr = r0 + rshift + i;
            if (gr < (long)N) {
                float v = acc[i] * ivr[i];
                dis[gr * DSTR + col] = fminf(fmaxf(v, -10.0f), 10.0f);
            }
        }
    }
}

// ---------------------------------------------------------------------------
// K2a: Sinkhorn iteration 1 (v0 = ones) + repack the row's own 5 class
// columns into compact dsel[N][8] for coalesced later passes.
// ---------------------------------------------------------------------------
__global__ __launch_bounds__(256) void sinkhorn_first_kernel(
        const float* __restrict__ dis,       // N x 64
        const int*   __restrict__ targets,
        float*       __restrict__ dsel,      // N x 8 compact
        float*       __restrict__ partials,  // [gridDim.x][50]
        int N) {
    __shared__ float lacc[NPX];
    for (int i = threadIdx.x; i < NPX; i += blockDim.x) lacc[i] = 0.f;
    __syncthreads();

    for (long n = (long)blockIdx.x * blockDim.x + threadIdx.x; n < N;
         n += (long)gridDim.x * blockDim.x) {
        int c = targets[n];
        const float* dr = dis + n * DSTR + c * PPC;
        float d0 = dr[0], d1 = dr[1], d2 = dr[2], d3 = dr[3], d4 = dr[4];

        float* pw = dsel + n * 8;
        *(float4*)pw = make_float4(d0, d1, d2, d3);
        pw[4] = d4;

        float s = d0 + d1 + d2 + d3 + d4;              // dot with v0 = ones
        float u = 1.0f / fmaxf(s, 1e-10f);
        atomicAdd(&lacc[c * PPC + 0], d0 * u);
        atomicAdd(&lacc[c * PPC + 1], d1 * u);
        atomicAdd(&lacc[c * PPC + 2], d2 * u);
        atomicAdd(&lacc[c * PPC + 3], d3 * u);
        atomicAdd(&lacc[c * PPC + 4], d4 * u);
    }
    __syncthreads();
    for (int i = threadIdx.x; i < NPX; i += blockDim.x)
        partials[(long)blockIdx.x * NPX + i] = lacc[i];
}

// ---------------------------------------------------------------------------
// K2b: Sinkhorn iterations 2..5 over the compact buffer.
// ---------------------------------------------------------------------------
__global__ __launch_bounds__(256) void sinkhorn_pass_kernel(
        const float* __restrict__ dsel,
        const int*   __restrict__ targets,
        const float* __restrict__ vin,       // 50 (stride-64 region)
        float*       __restrict__ partials,  // [gridDim.x][50]
        int N) {
    __shared__ float lacc[NPX];
    for (int i = threadIdx.x; i < NPX; i += blockDim.x) lacc[i] = 0.f;
    __syncthreads();

    for (long n = (long)blockIdx.x * blockDim.x + threadIdx.x; n < N;
         n += (long)gridDim.x * blockDim.x) {
        int c = targets[n];
        const float* dr = dsel + n * 8;
        float4 dv = *(const float4*)dr;
        float  d4 = dr[4];
        const float* vc = vin + c * PPC;
        float s = dv.x * vc[0] + dv.y * vc[1] + dv.z * vc[2] + dv.w * vc[3]
                + d4 * vc[4];
        float u = 1.0f / fmaxf(s, 1e-10f);
        atomicAdd(&lacc[c * PPC + 0], dv.x * u);
        atomicAdd(&lacc[c * PPC + 1], dv.y * u);
        atomicAdd(&lacc[c * PPC + 2], dv.z * u);
        atomicAdd(&lacc[c * PPC + 3], dv.w * u);
        atomicAdd(&lacc[c * PPC + 4], d4  * u);
    }
    __syncthreads();
    for (int i = threadIdx.x; i < NPX; i += blockDim.x)
        partials[(long)blockIdx.x * NPX + i] = lacc[i];
}

// K3: deterministic reduce -> v_out = 1/clip(colsum)
__global__ void v_reduce_kernel(const float* __restrict__ partials,
                                float* __restrict__ vout, int nb) {
    int t = threadIdx.x;
    if (t < NPX) {
        float s0 = 0.f, s1 = 0.f, s2 = 0.f, s3 = 0.f;
        for (int b = 0; b < nb; b += 4) {
            s0 += partials[(long)(b + 0) * NPX + t];
            s1 += partials[(long)(b + 1) * NPX + t];
            s2 += partials[(long)(b + 2) * NPX + t];
            s3 += partials[(long)(b + 3) * NPX + t];
        }
        float s = (s0 + s1) + (s2 + s3);
        vout[t] = 1.0f / fmaxf(s, 1e-10f);
    }
}

// ---------------------------------------------------------------------------
// K4: final per-row pass: Qrow, sum_j Q*logits, lse, count -> per-class sums
// ---------------------------------------------------------------------------
__global__ __launch_bounds__(256) void final_pass_kernel(
        const float* __restrict__ dsel,
        const int*   __restrict__ targets,
        const float* __restrict__ v4,        // v after iter 4 (for u5)
        const float* __restrict__ v5,        // v after iter 5 (in Q)
        float*       __restrict__ partials,  // [gridDim.x][40]
        int N) {
    __shared__ float lacc[4 * CLS];
    for (int i = threadIdx.x; i < 4 * CLS; i += blockDim.x) lacc[i] = 0.f;
    __syncthreads();

    for (long n = (long)blockIdx.x * blockDim.x + threadIdx.x; n < N;
         n += (long)gridDim.x * blockDim.x) {
        int c = targets[n];
        const float* dr  = dsel + n * 8;
        const float* vc4 = v4 + c * PPC;
        const float* vc5 = v5 + c * PPC;

        float4 dv = *(const float4*)dr;
        float d[PPC] = { dv.x, dv.y, dv.z, dv.w, dr[4] };

        float sdot = 0.f, mx = -1e30f, x[PPC];
#pragma unroll
        for (int j = 0; j < PPC; ++j) {
            sdot += d[j] * vc4[j];
            x[j] = d[j] * INV_T;             // dis/TEMP == dis/EPS
            mx = fmaxf(mx, x[j]);
        }
        float u = 1.0f / fmaxf(sdot, 1e-10f);

        float se = 0.f;
#pragma unroll
        for (int j = 0; j < PPC; ++j) se += expf(x[j] - mx);
        float lse1 = logf(se);

        float qs = 0.f, t1 = 0.f, s2 = 0.f;
#pragma unroll
        for (int j = 0; j < PPC; ++j) {
            float q = expf(x[j]) * vc5[j];   // exp(dis/EPS)*v
            float l = x[j] - mx - lse1;      // log_softmax
            qs += q;
            t1 += q * l;
            s2 += expf(l);
        }
        float lse = logf(s2);                // logsumexp(log_softmax) ~ 0

        atomicAdd(&lacc[4 * c + 0], u * qs); // Qsum contribution
        atomicAdd(&lacc[4 * c + 1], u * t1); // sum Q*logits
        atomicAdd(&lacc[4 * c + 2], lse);    // sum m*lse
        atomicAdd(&lacc[4 * c + 3], 1.0f);   // count
    }
    __syncthreads();
    for (int i = threadIdx.x; i < 4 * CLS; i += blockDim.x)
        partials[(long)blockIdx.x * (4 * CLS) + i] = lacc[i];
}

// ---------------------------------------------------------------------------
// K5: single block: reduce class sums -> mle; compute pcon from proxy gram;
// out[0] = (mle + pcon)/C
// ---------------------------------------------------------------------------
__global__ __launch_bounds__(256) void finalize_kernel(
        const float* __restrict__ proxy,
        const float* __restrict__ partials, int nb,
        float* __restrict__ out) {
    __shared__ float G[NPX * NPX];
    __shared__ float terms[CLS * CLS];
    __shared__ float sums[4 * CLS];

    // gram matrix 50x50 (proxy already unit-norm)
    for (int idx = threadIdx.x; idx < NPX * NPX; idx += blockDim.x) {
        int i = idx / NPX, j = idx % NPX;
        const float4* a = (const float4*)(proxy + (long)i * DIM);
        const float4* b = (const float4*)(proxy + (long)j * DIM);
        float s = 0.f;
        for (int k = 0; k < DIM / 4; ++k) {
            float4 xa = a[k], xb = b[k];
            s += xa.x * xb.x + xa.y * xb.y + xa.z * xb.z + xa.w * xb.w;
        }
        G[idx] = s;
    }

    int t = threadIdx.x;
    if (t < 4 * CLS) {                       // deterministic cross-block reduce
        float s = 0.f;
        for (int b = 0; b < nb; ++b) s += partials[(long)b * (4 * CLS) + t];
        sums[t] = s;
    }
    __syncthreads();

    if (t < CLS * CLS) {                     // one (c, co) pair per thread
        int c = t / CLS, co = t % CLS;
        float acc = 0.f;
        for (int i = 0; i < PPC; ++i) {
            float l[PPC], mx = -1e30f;
            for (int j = 0; j < PPC; ++j) {
                float g = G[(c * PPC + i) * NPX + co * PPC + j] * INV_T;
                g = fminf(fmaxf(g, -10.0f), 10.0f);
                l[j] = g;
                mx = fmaxf(mx, g);
            }
            float se = 0.f;
            for (int j = 0; j < PPC; ++j) se += expf(l[j] - mx);
            float lse1 = logf(se);
            float s2 = 0.f;
            for (int j = 0; j < PPC; ++j) s2 += expf(l[j] - mx - lse1);
            float lse = logf(s2);
            if (c == co) acc += -((l[i] - mx - lse1) - lse); // -(diag - lse)
            else         acc += lse;
        }
        terms[t] = acc * (1.0f / PPC);       // mean over P
    }
    __syncthreads();

    if (t == 0) {
        float pcon = 0.f;
        for (int c = 0; c < CLS; ++c) {
            pcon += terms[c * CLS + c];
            for (int co = 0; co < CLS; ++co)
                if (co != c) pcon += terms[c * CLS + co];
        }
        float mle = 0.f;
        for (int c = 0; c < CLS; ++c) {
            float Q = sums[4 * c + 0], T1 = sums[4 * c + 1];
            float T2 = sums[4 * c + 2], cnt = sums[4 * c + 3];
            if (cnt > 0.f) mle += -(T1 / Q - T2) / fmaxf(cnt, 1.0f);
        }
        out[0] = (mle + pcon) / (float)CLS;
    }
}

// ---------------------------------------------------------------------------
extern "C" void kernel_launch(void* const* d_in, const int* in_sizes, int n_in,
                              void* d_out, int out_size, void* d_ws, size_t ws_size,
                              hipStream_t stream) {
    const float* features = (const float*)d_in[0];
    const float* proxy    = (const float*)d_in[1];
    const int*   targets  = (const int*)d_in[2];
    float*       out      = (float*)d_out;

    const int N = in_sizes[0] / DIM;

    // workspace layout (floats)
    float* ws   = (float*)d_ws;
    size_t off  = 0;
    float* dis  = ws + off; off += (size_t)N * DSTR;         // N x 64
    float* pp   = ws + off; off += (size_t)NPAD * DIM;       // padded proxy
    float* v    = ws + off; off += 6 * 64;                   // v[1..5], stride 64
    float* dsel = ws + off; off += (size_t)N * 8;            // compact rows
    float* prt1 = ws + off; off += (size_t)NB_RED * NPX;     // sinkhorn partials
    float* prt2 = ws + off; off += (size_t)NB_RED * 4 * CLS; // final partials

    setup_kernel<<<NPAD, 256, 0, stream>>>(proxy, pp);

    int tiles = (N + 15) / 16;
    dis_wmma_kernel<<<tiles, 128, 0, stream>>>(features, pp, dis, N);

    // iteration 1 (v0 = ones) + repack
    sinkhorn_first_kernel<<<NB_RED, 256, 0, stream>>>(dis, targets, dsel, prt1, N);
    v_reduce_kernel<<<1, 64, 0, stream>>>(prt1, v + 1 * 64, NB_RED);

    // iterations 2..5
    for (int it = 2; it <= 5; ++it) {
        sinkhorn_pass_kernel<<<NB_RED, 256, 0, stream>>>(
            dsel, targets, v + (it - 1) * 64, prt1, N);
        v_reduce_kernel<<<1, 64, 0, stream>>>(prt1, v + it * 64, NB_RED);
    }

    final_pass_kernel<<<NB_RED, 256, 0, stream>>>(
        dsel, targets, v + 4 * 64, v + 5 * 64, prt2, N);

    finalize_kernel<<<1, 256, 0, stream>>>(proxy, prt2, NB_RED, out);
}